// QuantileLoss_73512660239006
// MI455X (gfx1250) — compile-verified
//
#include <hip/hip_runtime.h>

typedef __attribute__((ext_vector_type(2))) float v2f;
typedef __attribute__((ext_vector_type(8))) float v8f;

// ---------------------------------------------------------------------------
// Per-row quantile loss (transcribed from the reference):
//   main  = (p0-t0)^2 + (p1-t1)^2 + (p2-t2)^2
//   lower = p3>p2 ? 4*(p4-t2) : (p3>0.95*t2 ? 0 : (p3-0.95*t2)^2)
//   upper = p4<p2 ? 4*(p4-t2) : (p4<1.05*t2 ? 0 : (p4-1.05*t2)^2)
// ---------------------------------------------------------------------------
__device__ __forceinline__ float row_loss(float p0, float p1, float p2,
                                          float p3, float p4,
                                          float t0, float t1, float t2) {
    float d0 = p0 - t0, d1 = p1 - t1, d2 = p2 - t2;
    float main_mse = d0 * d0 + d1 * d1 + d2 * d2;
    float tl = 0.95f * t2;
    float tu = 1.05f * t2;
    float lin = (p4 - t2) * 4.0f;
    float dl = p3 - tl;
    float du = p4 - tu;
    float lower = (p3 > p2) ? lin : ((p3 > tl) ? 0.0f : dl * dl);
    float upper = (p4 < p2) ? lin : ((p4 < tu) ? 0.0f : du * du);
    return main_mse + lower + upper;
}

// ---------------------------------------------------------------------------
// Wave32 cross-lane sum via V_WMMA_F32_16X16X4_F32.
// B-matrix (4x16, 2 VGPRs): VGPR0 = this lane's partial, VGPR1 = 0.
//   -> two of the four K-rows of B hold {s_0..s_15} (lanes 0-15) and
//      {s_16..s_31} (lanes 16-31) in columns 0..15; the rest are zero.
// A-matrix = all ones (16x4), C = 0:
//   D[m][n] = sum_k B[k][n] = s_n + s_{n+16}   (independent of K-row mapping)
// Every row of D is identical, so VGPR0 of D gives lane l the value
// r[l & 15] = s_{l&15} + s_{(l&15)+16}.  Four xor-shuffles (1,2,4,8) within
// the 16-lane group complete the 32-lane sum; all lanes end with the total.
// EXEC must be all ones at the call site (full wave, no divergence).
// ---------------------------------------------------------------------------
__device__ __forceinline__ float wave_sum32(float s) {
    v2f a;
    a.x = 1.0f;
    a.y = 1.0f;
    v2f b;
    b.x = s;
    b.y = 0.0f;
    v8f c = {};
    c = __builtin_amdgcn_wmma_f32_16x16x4_f32(
        /*neg_a=*/false, a, /*neg_b=*/false, b,
        /*c_mod=*/(short)0, c, /*reuse_a=*/false, /*reuse_b=*/false);
    float r = c[0];
    r += __shfl_xor(r, 1, 32);
    r += __shfl_xor(r, 2, 32);
    r += __shfl_xor(r, 4, 32);
    r += __shfl_xor(r, 8, 32);
    return r;
}

// ---------------------------------------------------------------------------
// Stage 1: each thread consumes 4 rows with fully aligned float4 (b128) loads:
//   preds:  rows [4t,4t+4) = 20 floats = float4[5t .. 5t+5)   (80 B, 16B align)
//   target: rows [4t,4t+4) = 12 floats = float4[3t .. 3t+3)   (48 B, 16B align)
// One partial per block written to d_ws (no atomics -> deterministic replay).
// ---------------------------------------------------------------------------
__global__ void ql_partial_kernel(const float* __restrict__ preds,
                                  const float* __restrict__ target,
                                  float* __restrict__ partials,
                                  int n_rows) {
    const int t = blockIdx.x * blockDim.x + threadIdx.x;  // 4 rows per thread
    const int r0 = t * 4;

    float s = 0.0f;
    if (r0 + 3 < n_rows) {
        const float4* pv = (const float4*)preds;
        const float4* tv = (const float4*)target;
        float4 P0 = pv[5 * t + 0];
        float4 P1 = pv[5 * t + 1];
        float4 P2 = pv[5 * t + 2];
        float4 P3 = pv[5 * t + 3];
        float4 P4 = pv[5 * t + 4];
        float4 T0 = tv[3 * t + 0];
        float4 T1 = tv[3 * t + 1];
        float4 T2 = tv[3 * t + 2];
        s += row_loss(P0.x, P0.y, P0.z, P0.w, P1.x, T0.x, T0.y, T0.z);
        s += row_loss(P1.y, P1.z, P1.w, P2.x, P2.y, T0.w, T1.x, T1.y);
        s += row_loss(P2.z, P2.w, P3.x, P3.y, P3.z, T1.z, T1.w, T2.x);
        s += row_loss(P3.w, P4.x, P4.y, P4.z, P4.w, T2.y, T2.z, T2.w);
    } else {
        // Safe scalar tail (never taken for N = 4194304, kept for generality).
        for (int r = r0; r < n_rows && r < r0 + 4; ++r) {
            s += row_loss(preds[r * 5 + 0], preds[r * 5 + 1], preds[r * 5 + 2],
                          preds[r * 5 + 3], preds[r * 5 + 4],
                          target[r * 3 + 0], target[r * 3 + 1], target[r * 3 + 2]);
        }
    }

    // All 256 threads reach here (uniform control flow) -> EXEC all ones.
    float w = wave_sum32(s);

    __shared__ float lds[8];
    const int lane = threadIdx.x & 31;
    const int wid = threadIdx.x >> 5;
    if (lane == 0) lds[wid] = w;
    __syncthreads();
    if (threadIdx.x == 0) {
        float bsum = 0.0f;
        for (int i = 0; i < 8; ++i) bsum += lds[i];
        partials[blockIdx.x] = bsum;
    }
}

// ---------------------------------------------------------------------------
// Stage 2: single 256-thread block reduces the per-block partials and writes
// the mean (divide by 5*N total loss elements).
// ---------------------------------------------------------------------------
__global__ void ql_finalize_kernel(const float* __restrict__ partials,
                                   float* __restrict__ out,
                                   int n_partials,
                                   float inv_denom) {
    float s = 0.0f;
    for (int i = threadIdx.x; i < n_partials; i += blockDim.x) s += partials[i];

    float w = wave_sum32(s);

    __shared__ float lds[8];
    const int lane = threadIdx.x & 31;
    const int wid = threadIdx.x >> 5;
    if (lane == 0) lds[wid] = w;
    __syncthreads();
    if (threadIdx.x == 0) {
        float total = 0.0f;
        for (int i = 0; i < 8; ++i) total += lds[i];
        out[0] = total * inv_denom;
    }
}

extern "C" void kernel_launch(void* const* d_in, const int* in_sizes, int n_in,
                              void* d_out, int out_size, void* d_ws, size_t ws_size,
                              hipStream_t stream) {
    const float* preds = (const float*)d_in[0];   // (N,5) f32
    const float* target = (const float*)d_in[1];  // (N,3) f32
    float* partials = (float*)d_ws;
    float* out = (float*)d_out;

    const int n_rows = in_sizes[0] / 5;                 // N = 4194304
    const int n_threads = (n_rows + 3) / 4;             // 4 rows / thread
    const int block = 256;                              // 8 wave32s
    const int grid = (n_threads + block - 1) / block;   // 4096 blocks

    ql_partial_kernel<<<grid, block, 0, stream>>>(preds, target, partials, n_rows);

    const float inv_denom = 1.0f / (5.0f * (float)n_rows);
    ql_finalize_kernel<<<1, block, 0, stream>>>(partials, out, grid, inv_denom);
}